// Upsampling3D_53790170415479
// MI455X (gfx1250) — compile-verified
//
#include <hip/hip_runtime.h>
#include <cstdint>

#define CCH 8
#define FFT 32
#define CF  256   // CCH * FFT floats per node (1 KB)

// ---------------- zero the workspace accumulators ----------------
__global__ __launch_bounds__(256) void uz_zero(float4* __restrict__ p, long long n4) {
  long long i = (long long)blockIdx.x * blockDim.x + threadIdx.x;
  if (i < n4) p[i] = make_float4(0.f, 0.f, 0.f, 0.f);
}

// ---- issue async global->LDS staging of one 1KB feature row (2 x b128/lane)
__device__ __forceinline__ void uz_stage(const float* __restrict__ gsrc,
                                         float* __restrict__ ldst) {
  const uint32_t la = (uint32_t)(uintptr_t)ldst;   // low 32b = LDS byte offset
  const uint64_t ga = (uint64_t)(uintptr_t)gsrc;
  asm volatile("global_load_async_to_lds_b128 %0, %1, off"
               :: "v"(la), "v"(ga) : "memory");
  asm volatile("global_load_async_to_lds_b128 %0, %1, off offset:512"
               :: "v"(la), "v"(ga) : "memory");
}

// ---- accumulate one staged edge row into the destination accumulators
__device__ __forceinline__ void uz_accum(const float* __restrict__ rowbuf,
                                         int lane, float inv,
                                         float* __restrict__ aggp,
                                         float* __restrict__ normp) {
  float v[CCH];
  unsigned zmask = 0u;                       // wave-uniform per-channel any(v==0)
  #pragma unroll
  for (int c = 0; c < CCH; ++c) {
    v[c] = rowbuf[c * FFT + lane];           // lane == feature index (F==32)
#if __has_builtin(__builtin_amdgcn_ballot_w32)
    if (__builtin_amdgcn_ballot_w32(v[c] == 0.0f) != 0u) zmask |= (1u << c);
#else
    if (__any(v[c] == 0.0f)) zmask |= (1u << c);
#endif
  }
  #pragma unroll
  for (int c = 0; c < CCH; ++c) {
    const float wt = ((zmask >> c) & 1u) ? 0.0f : inv;
    atomicAdd(&aggp[c * FFT + lane], v[c] * wt);     // global_atomic_add_f32
  }
  if (lane < CCH) {                          // one atomic instr, 8 active lanes
    const float wt = ((zmask >> lane) & 1u) ? 0.0f : inv;
    atomicAdd(&normp[lane], wt);
  }
}

// ---------------- edge scatter (IDW accumulate) ----------------
// One wave32 per TWO edges. Both 1KB source rows are staged with async
// global->LDS loads; ASYNCcnt's in-order completion lets us process edge 0
// (s_wait_asynccnt 2) while edge 1's row is still in flight.
__global__ __launch_bounds__(256) void uz_edges(
    const float* __restrict__ feat,   // [n_sub, C, F]
    const float* __restrict__ edge_w, // [E]
    const int*   __restrict__ srcs,   // [E], in [0, n_sub)
    const int*   __restrict__ dsts,   // [E], in [n_sub, n_full)
    float*       __restrict__ agg,    // [n_up, C, F]
    float*       __restrict__ norm,   // [n_up, C]
    int E, int n_sub, int n_full)
{
  __shared__ __align__(16) float rows[8][2][CF];   // 16 KB: 2 row buffers/wave
  const int lane = threadIdx.x & 31;
  const int wid  = threadIdx.x >> 5;
  const int e0 = (blockIdx.x * 8 + wid) * 2;       // wave-uniform
  if (e0 >= E) return;
  const bool has1 = (e0 + 1) < E;

  const int   s0 = srcs[e0];
  const int   d0 = dsts[e0];
  const float w0 = edge_w[e0];
  const int   s1 = has1 ? srcs[e0 + 1]   : 0;
  const int   d1 = has1 ? dsts[e0 + 1]   : 0;
  const float w1 = has1 ? edge_w[e0 + 1] : 1.0f;

  uz_stage(feat + (size_t)s0 * CF + lane * 4, &rows[wid][0][lane * 4]);
  if (has1)
    uz_stage(feat + (size_t)s1 * CF + lane * 4, &rows[wid][1][lane * 4]);

  if (has1) {   // first row's 2 loads complete in-order -> ASYNCcnt <= 2
    asm volatile("s_wait_asynccnt 2" ::: "memory");
  } else {
    asm volatile("s_wait_asynccnt 0" ::: "memory");
  }

  if (d0 >= n_sub && d0 < n_full) {
    const size_t m0 = (size_t)(d0 - n_sub);
    uz_accum(rows[wid][0], lane, 1.0f / (w0 * w0),
             agg + m0 * CF, norm + m0 * CCH);
  }

  if (has1) {
    asm volatile("s_wait_asynccnt 0" ::: "memory");
    if (d1 >= n_sub && d1 < n_full) {
      const size_t m1 = (size_t)(d1 - n_sub);
      uz_accum(rows[wid][1], lane, 1.0f / (w1 * w1),
               agg + m1 * CF, norm + m1 * CCH);
    }
  }
}

// ---------------- finalize: normalize + write output ----------------
__global__ __launch_bounds__(256) void uz_final(
    const float* __restrict__ feat, const int* __restrict__ fp_idx,
    const float* __restrict__ agg,  const float* __restrict__ norm,
    float* __restrict__ out, int n_sub, int n_full)
{
  const long long i4 = (long long)blockIdx.x * blockDim.x + threadIdx.x;
  const long long total4 = (long long)n_full * (CF / 4);
  if (i4 >= total4) return;
  const int node = (int)(i4 >> 6);     // 64 float4 per node
  const int r4   = (int)(i4 & 63);
  if (node < n_sub) {
    const float4 v = ((const float4*)feat)[i4];
    const int dn = fp_idx[node];
    ((float4*)out)[(size_t)dn * 64 + r4] = v;
  } else {
    const size_t m = (size_t)(node - n_sub);
    const int c = r4 >> 3;                         // 8 float4 per channel
    const float nv = norm[m * CCH + c];
    const float sn = nv > 0.0f ? nv : 1.0f;
    const float4 a = ((const float4*)agg)[m * 64 + r4];
    float4 r; r.x = a.x / sn; r.y = a.y / sn; r.z = a.z / sn; r.w = a.w / sn;
    ((float4*)out)[i4] = r;
  }
}

extern "C" void kernel_launch(void* const* d_in, const int* in_sizes, int n_in,
                              void* d_out, int out_size, void* d_ws, size_t ws_size,
                              hipStream_t stream) {
  (void)n_in; (void)ws_size;
  const float* feat   = (const float*)d_in[0];  // [n_sub, 8, 32] f32
  const float* edge_w = (const float*)d_in[1];  // [E] f32
  const int*   fp_idx = (const int*)d_in[2];    // [n_sub] i32
  const int*   srcs   = (const int*)d_in[3];    // [E] i32
  const int*   dsts   = (const int*)d_in[4];    // [E] i32
  float*       out    = (float*)d_out;          // [n_full, 8, 32] f32

  const int E      = in_sizes[1];
  const int n_sub  = in_sizes[2];
  const int n_full = out_size / CF;
  const int n_up   = n_full - n_sub;

  float* agg  = (float*)d_ws;                      // n_up * 256 floats
  float* norm = agg + (size_t)n_up * CF;           // n_up * 8 floats

  // 1) zero accumulators (every call: ws is not re-poisoned between replays)
  const long long z4 = (long long)n_up * (CF + CCH) / 4;
  const unsigned zb = (unsigned)((z4 + 255) / 256);
  uz_zero<<<zb, 256, 0, stream>>>((float4*)d_ws, z4);

  // 2) edge scatter: 8 waves/block, 2 edges/wave (async double-buffered)
  const unsigned eb = (unsigned)((E + 15) / 16);
  uz_edges<<<eb, 256, 0, stream>>>(feat, edge_w, srcs, dsts, agg, norm,
                                   E, n_sub, n_full);

  // 3) normalize + emit (float4 stores)
  const long long t4 = (long long)n_full * (CF / 4);
  const unsigned fb = (unsigned)((t4 + 255) / 256);
  uz_final<<<fb, 256, 0, stream>>>(feat, fp_idx, agg, norm, out, n_sub, n_full);
}